// RegNet_v2_42219528520117
// MI455X (gfx1250) — compile-verified
//
#include <hip/hip_runtime.h>
#include <math.h>

typedef float v2f __attribute__((ext_vector_type(2)));
typedef float v8f __attribute__((ext_vector_type(8)));

#define BN_EPS 1e-5f

// ---------------------------------------------------------------------------
// Farthest point sampling: one block per batch element. dist[] lives in LDS.
// Matches jax reference: emits carry `far` each step (far_0 = 0), argmax with
// first-occurrence tie-break.
// ---------------------------------------------------------------------------
#define FPS_THREADS 256

__global__ void fps_kernel(const float* __restrict__ xyz, int N, int S,
                           int* __restrict__ fidx) {
    __shared__ float sdist[8192];
    __shared__ float redv[FPS_THREADS];
    __shared__ int   redi[FPS_THREADS];
    __shared__ int   s_far;

    const int b = blockIdx.x;
    const int t = threadIdx.x;
    const float* X = xyz + (size_t)b * N * 3;

    for (int i = t; i < N; i += FPS_THREADS) sdist[i] = 1e10f;
    if (t == 0) s_far = 0;
    __syncthreads();

    for (int it = 0; it < S; ++it) {
        const int far = s_far;
        if (t == 0) fidx[b * S + it] = far;

        const float cx = X[far * 3 + 0];
        const float cy = X[far * 3 + 1];
        const float cz = X[far * 3 + 2];

        float bestv = -1e30f;
        int   besti = 0x7fffffff;
        for (int i = t; i < N; i += FPS_THREADS) {
            const float dx = X[i * 3 + 0] - cx;
            const float dy = X[i * 3 + 1] - cy;
            const float dz = X[i * 3 + 2] - cz;
            const float d  = dx * dx + dy * dy + dz * dz;
            float nd = sdist[i];
            nd = (d < nd) ? d : nd;
            sdist[i] = nd;
            if (nd > bestv) { bestv = nd; besti = i; }  // strict > keeps lowest i
        }
        redv[t] = bestv;
        redi[t] = besti;
        __syncthreads();

        for (int sft = FPS_THREADS / 2; sft > 0; sft >>= 1) {
            if (t < sft) {
                const float v2 = redv[t + sft];
                const int   i2 = redi[t + sft];
                if (v2 > redv[t] || (v2 == redv[t] && i2 < redi[t])) {
                    redv[t] = v2;
                    redi[t] = i2;
                }
            }
            __syncthreads();
        }
        if (t == 0) s_far = redi[0];
        __syncthreads();
    }
}

// ---------------------------------------------------------------------------
// Gather sampled centers: new_xyz[b,s,:] = xyz[b, fidx[b,s], :]
// ---------------------------------------------------------------------------
__global__ void gather_centers_kernel(const float* __restrict__ xyz,
                                      const int* __restrict__ fidx,
                                      float* __restrict__ new_xyz,
                                      int N, int S, int total /* = B*S */) {
    const int i = blockIdx.x * blockDim.x + threadIdx.x;
    if (i >= total) return;
    const int b   = i / S;
    const int idx = fidx[i];
    const float* p = xyz + ((size_t)b * N + idx) * 3;
    new_xyz[i * 3 + 0] = p[0];
    new_xyz[i * 3 + 1] = p[1];
    new_xyz[i * 3 + 2] = p[2];
}

// ---------------------------------------------------------------------------
// Ball query: first `ns` in-radius indices (ascending), padded with first hit.
// ---------------------------------------------------------------------------
__global__ void ball_query_kernel(const float* __restrict__ xyz,
                                  const float* __restrict__ new_xyz,
                                  int* __restrict__ nidx,
                                  int N, int S, int ns, float r2,
                                  int total /* = B*S */) {
    const int i = blockIdx.x * blockDim.x + threadIdx.x;
    if (i >= total) return;
    const int b = i / S;
    const float cx = new_xyz[i * 3 + 0];
    const float cy = new_xyz[i * 3 + 1];
    const float cz = new_xyz[i * 3 + 2];
    const float* X = xyz + (size_t)b * N * 3;

    const int base = i * ns;
    int cnt = 0, first = -1;
    for (int j = 0; j < N && cnt < ns; ++j) {
        const float dx = X[j * 3 + 0] - cx;
        const float dy = X[j * 3 + 1] - cy;
        const float dz = X[j * 3 + 2] - cz;
        const float sq = dx * dx + dy * dy + dz * dz;
        if (sq <= r2) {
            if (first < 0) first = j;
            nidx[base + cnt] = j;
            ++cnt;
        }
    }
    if (first < 0) first = N - 1;  // unreachable (center is in the set), safe clamp
    for (; cnt < ns; ++cnt) nidx[base + cnt] = first;
}

// ---------------------------------------------------------------------------
// Group: out[b,s,k,:] = concat(xyz[nidx]-center, feats[nidx])
// ---------------------------------------------------------------------------
__global__ void group_kernel(const float* __restrict__ xyz,
                             const float* __restrict__ feats,
                             const float* __restrict__ new_xyz,
                             const int* __restrict__ nidx,
                             float* __restrict__ out,
                             int N, int S, int ns, int Cf,
                             int total /* = B*S*ns */) {
    const int i = blockIdx.x * blockDim.x + threadIdx.x;
    if (i >= total) return;
    const int bs  = i / ns;
    const int b   = bs / S;
    const int idx = nidx[i];
    const int Cin = 3 + Cf;

    const float* p = xyz + ((size_t)b * N + idx) * 3;
    const float* c = new_xyz + (size_t)bs * 3;
    float* o = out + (size_t)i * Cin;
    o[0] = p[0] - c[0];
    o[1] = p[1] - c[1];
    o[2] = p[2] - c[2];
    const float* f = feats + ((size_t)b * N + idx) * Cf;
    for (int cc = 0; cc < Cf; ++cc) o[3 + cc] = f[cc];
}

// ---------------------------------------------------------------------------
// GEMM + bias via CDNA5 fp32 WMMA: Y[R,COUT] = X[R,CIN] @ W[CIN,COUT] + b
// Compile-time CIN/COUT; each wave owns 4 row-tiles x 1 col-tile so the B
// (weight) fragment is reused across 4 WMMAs per K-step. R is guaranteed a
// multiple of 64 by the launcher, so there are no row guards. Ragged K and
// the COUT==8 case use clamped in-bounds addresses + 0/1 multiplicative
// masks: loads stay unconditional (no exec-mask branching).
// Fragment layouts per CDNA5 ISA 7.12.2 (wave32):
//   A(16x4):  lane L -> row L%16, a[0]=K(k0+2*(L>=16)), a[1]=K(..+1)
//   B(4x16):  lane L -> col L%16, b[0]=K(k0+2*(L>=16)), b[1]=K(..+1)
//   D(16x16): vgpr r -> row r+8*(L>=16), col L%16
// ---------------------------------------------------------------------------
template <int CIN, int COUT>
__global__ void gemm_bias_wmma_kernel(const float* __restrict__ X,
                                      const float* __restrict__ W,
                                      const float* __restrict__ bias,
                                      float* __restrict__ Y, int R) {
    constexpr int RT       = 4;                  // row tiles per wave
    constexpr int colTiles = (COUT + 15) / 16;
    constexpr bool colGuard = (COUT % 16) != 0;  // only COUT==8 here

    const int wavesPerBlock = blockDim.x >> 5;
    const int wid  = blockIdx.x * wavesPerBlock + (threadIdx.x >> 5);
    const int lane = threadIdx.x & 31;
    const int rowGroup = wid / colTiles;
    const int colTile  = wid % colTiles;
    const int row0 = rowGroup * (16 * RT);
    if (row0 >= R) return;  // wave-uniform

    const int halfSel = lane >> 4;   // 0 | 1
    const int l16     = lane & 15;
    const int colB    = colTile * 16 + l16;
    const int colC    = colGuard ? (colB < COUT ? colB : COUT - 1) : colB;
    const float cmask = (!colGuard || colB < COUT) ? 1.f : 0.f;

    v8f acc[RT];
#pragma unroll
    for (int rt = 0; rt < RT; ++rt)
        acc[rt] = (v8f){0.f, 0.f, 0.f, 0.f, 0.f, 0.f, 0.f, 0.f};

#pragma unroll
    for (int k0 = 0; k0 < CIN; k0 += 4) {
        const int   kk  = k0 + 2 * halfSel;
        const int   kc0 = (kk     < CIN) ? kk     : 0;   // clamped, in-bounds
        const int   kc1 = (kk + 1 < CIN) ? kk + 1 : 0;
        const float m0  = (kk     < CIN) ? 1.f : 0.f;    // folds to 1 in body
        const float m1  = (kk + 1 < CIN) ? 1.f : 0.f;

        v2f b;
        b.x = W[(size_t)kc0 * COUT + colC] * (m0 * cmask);
        b.y = W[(size_t)kc1 * COUT + colC] * (m1 * cmask);

#pragma unroll
        for (int rt = 0; rt < RT; ++rt) {
            const int rowA = row0 + rt * 16 + l16;
            v2f a;
            a.x = X[(size_t)rowA * CIN + kc0] * m0;
            a.y = X[(size_t)rowA * CIN + kc1] * m1;
            acc[rt] = __builtin_amdgcn_wmma_f32_16x16x4_f32(
                /*neg_a=*/false, a, /*neg_b=*/false, b,
                /*c_mod=*/(short)0, acc[rt], /*reuse_a=*/false, /*reuse_b=*/false);
        }
    }

    const float bv = bias[colC];
#pragma unroll
    for (int rt = 0; rt < RT; ++rt) {
#pragma unroll
        for (int r = 0; r < 8; ++r) {
            const int row = row0 + rt * 16 + r + 8 * halfSel;
            if (!colGuard || colB < COUT)
                Y[(size_t)row * COUT + colB] = acc[rt][r] + bv;
        }
    }
}

// ---------------------------------------------------------------------------
// BatchNorm over activations (mean/var over all rows), then ReLU.
// ---------------------------------------------------------------------------
__global__ void zero_kernel(float* __restrict__ p, int n) {
    const int i = blockIdx.x * blockDim.x + threadIdx.x;
    if (i < n) p[i] = 0.f;
}

__global__ void bn_stats_kernel(const float* __restrict__ X,
                                float* __restrict__ stats,  // [2*C]: sum, sumsq
                                int R, int C) {
    __shared__ float ssum[128];
    __shared__ float ssq[128];
    for (int c = threadIdx.x; c < C; c += blockDim.x) { ssum[c] = 0.f; ssq[c] = 0.f; }
    __syncthreads();

    const size_t total  = (size_t)R * C;
    const size_t stride = (size_t)gridDim.x * blockDim.x;
    for (size_t i = (size_t)blockIdx.x * blockDim.x + threadIdx.x; i < total; i += stride) {
        const float v = X[i];
        const int   c = (int)(i % (size_t)C);
        atomicAdd(&ssum[c], v);       // ds_add_f32
        atomicAdd(&ssq[c],  v * v);
    }
    __syncthreads();
    for (int c = threadIdx.x; c < C; c += blockDim.x) {
        atomicAdd(&stats[c],     ssum[c]);
        atomicAdd(&stats[C + c], ssq[c]);
    }
}

__global__ void bn_apply_relu_kernel(float* __restrict__ X,
                                     const float* __restrict__ stats,
                                     const float* __restrict__ g,
                                     const float* __restrict__ be,
                                     int R, int C) {
    const size_t i = (size_t)blockIdx.x * blockDim.x + threadIdx.x;
    if (i >= (size_t)R * C) return;
    const int   c    = (int)(i % (size_t)C);
    const float inv  = 1.f / (float)R;
    const float mean = stats[c] * inv;
    const float var  = stats[C + c] * inv - mean * mean;
    const float v    = (X[i] - mean) * rsqrtf(var + BN_EPS) * g[c] + be[c];
    X[i] = v > 0.f ? v : 0.f;
}

// ---------------------------------------------------------------------------
// Max-pool over neighborhood dim: out[bs,c] = max_k x[bs,k,c]
// ---------------------------------------------------------------------------
__global__ void maxpool_kernel(const float* __restrict__ X,
                               float* __restrict__ out,
                               int BS, int ns, int C) {
    const int i = blockIdx.x * blockDim.x + threadIdx.x;
    if (i >= BS * C) return;
    const int c  = i % C;
    const int bs = i / C;
    const float* p = X + (size_t)bs * ns * C + c;
    float m = p[0];
    for (int k = 1; k < ns; ++k) m = fmaxf(m, p[(size_t)k * C]);
    out[i] = m;
}

// ---------------------------------------------------------------------------
// Host orchestration
// ---------------------------------------------------------------------------
static inline int cdiv(int a, int b) { return (a + b - 1) / b; }

static void launch_gemm(const float* X, const float* W, const float* b, float* Y,
                        int R, int Cin, int Cout, hipStream_t stream) {
    // R is a multiple of 64 for every layer in this network.
#define GEMM_CASE(CI, CO)                                                        \
    if (Cin == CI && Cout == CO) {                                               \
        const int totalWaves = (R / 64) * (((CO) + 15) / 16);                    \
        gemm_bias_wmma_kernel<CI, CO>                                            \
            <<<cdiv(totalWaves, 4), 128, 0, stream>>>(X, W, b, Y, R);            \
        return;                                                                  \
    }
    GEMM_CASE(6, 8)
    GEMM_CASE(8, 16)
    GEMM_CASE(16, 32)
    GEMM_CASE(35, 32)
    GEMM_CASE(32, 32)
    GEMM_CASE(32, 64)
    GEMM_CASE(67, 64)
    GEMM_CASE(64, 64)
    GEMM_CASE(64, 128)
    GEMM_CASE(131, 128)
    GEMM_CASE(128, 128)
#undef GEMM_CASE
}

extern "C" void kernel_launch(void* const* d_in, const int* in_sizes, int n_in,
                              void* d_out, int out_size, void* d_ws, size_t ws_size,
                              hipStream_t stream) {
    (void)in_sizes; (void)n_in; (void)out_size; (void)ws_size;

    const int   B = 2;
    const int   npointA[4]  = {2048, 1024, 256, 64};
    const float radiusA[4]  = {0.5f, 0.5f, 1.0f, 2.0f};
    const int   nsampleA[4] = {32, 16, 16, 16};
    const int   chsA[4][4]  = {{6, 8, 16, 32},
                               {35, 32, 32, 64},
                               {67, 64, 64, 128},
                               {131, 128, 128, 128}};

    const float* lidar = (const float*)d_in[0];

    // workspace carve-up (floats)
    float* ws = (float*)d_ws;
    size_t off = 0;
    auto alloc = [&](size_t n) { float* p = ws + off; off += n; return p; };
    float* nx[2] = {alloc(2 * 2048 * 3), alloc(2 * 2048 * 3)};   // new_xyz ping-pong
    float* fe[2] = {alloc(131072), alloc(131072)};               // feats ping-pong
    int*   fidx  = (int*)alloc(2 * 2048);
    int*   nidx  = (int*)alloc(2 * 2048 * 32);
    float* act0  = alloc(4194304);
    float* act1  = alloc(4194304);
    float* stats = alloc(256);

    // layer-1 input features: zeros (B, 8192, 3)
    {
        const int n = B * 8192 * 3;
        zero_kernel<<<cdiv(n, 256), 256, 0, stream>>>(fe[0], n);
    }

    const float* cur_xyz   = lidar;
    const float* cur_feats = fe[0];
    int N = 8192, Cf = 3;

    for (int l = 0; l < 4; ++l) {
        const int   S   = npointA[l];
        const int   ns  = nsampleA[l];
        const float r2  = radiusA[l] * radiusA[l];
        const int*  chs = chsA[l];
        const int   BS  = B * S;
        const int   R   = BS * ns;   // multiple of 64 for all layers

        // 1) FPS
        fps_kernel<<<B, FPS_THREADS, 0, stream>>>(cur_xyz, N, S, fidx);
        // 2) gather centers
        gather_centers_kernel<<<cdiv(BS, 256), 256, 0, stream>>>(cur_xyz, fidx, nx[l & 1], N, S, BS);
        // 3) ball query
        ball_query_kernel<<<cdiv(BS, 256), 256, 0, stream>>>(cur_xyz, nx[l & 1], nidx, N, S, ns, r2, BS);
        // 4) group + concat
        group_kernel<<<cdiv(R, 256), 256, 0, stream>>>(cur_xyz, cur_feats, nx[l & 1], nidx,
                                                       act0, N, S, ns, Cf, R);

        // 5) MLP: 3 x (WMMA-GEMM + bias -> BN(activation stats) -> ReLU)
        float* xin  = act0;
        float* xout = act1;
        for (int sub = 0; sub < 3; ++sub) {
            const int Cin  = chs[sub];
            const int Cout = chs[sub + 1];
            const int pidx = 1 + l * 12 + sub * 4;
            const float* Wp  = (const float*)d_in[pidx + 0];
            const float* bp  = (const float*)d_in[pidx + 1];
            const float* gp  = (const float*)d_in[pidx + 2];
            const float* bep = (const float*)d_in[pidx + 3];

            launch_gemm(xin, Wp, bp, xout, R, Cin, Cout, stream);

            zero_kernel<<<1, 256, 0, stream>>>(stats, 2 * Cout);
            {
                const size_t total = (size_t)R * Cout;
                int blocks = (int)((total + 255) / 256);
                if (blocks > 512) blocks = 512;
                bn_stats_kernel<<<blocks, 256, 0, stream>>>(xout, stats, R, Cout);
            }
            {
                const size_t total = (size_t)R * Cout;
                bn_apply_relu_kernel<<<(int)((total + 255) / 256), 256, 0, stream>>>(
                    xout, stats, gp, bep, R, Cout);
            }
            float* t = xin; xin = xout; xout = t;  // result now in xin
        }

        // 6) max-pool over neighborhood -> next-layer features (or final output)
        const int Cl = chs[3];
        float* pooled = (l == 3) ? (float*)d_out : fe[(l + 1) & 1];
        maxpool_kernel<<<cdiv(BS * Cl, 256), 256, 0, stream>>>(xin, pooled, BS, ns, Cl);

        // advance
        cur_xyz   = nx[l & 1];
        cur_feats = pooled;
        N  = S;
        Cf = Cl;
    }
}